// Softmax_Loss_19189913879294
// MI455X (gfx1250) — compile-verified
//
#include <hip/hip_runtime.h>
#include <math.h>

// Gaussian-smoothed label loss:
//   loss = -(1/B) * sum_i [ (sum_{j valid} w_j * log(outputs[i, y_i+j-3])) / Z_i ]
//   Z_i  = sum_{j valid} outputs[i, y_i+j-3]
//   w_j  = exp(-0.5*((j-3)/4)^2) / (sqrt(2*pi)*4)       (data-independent)
//
// Memory-bound gather (28B used per 400B row): NT-hinted scalar loads,
// v_rcp_f32 instead of IEEE div, wave32 shuffle reduction + tiny LDS step,
// deterministic two-stage sum (no atomics -> bit-stable across replays).

#define C_NUM 100
#define BAND 3
#define WIN 7
#define BLOCK 256
#define GRID 2048
#define WAVES_PER_BLOCK (BLOCK / 32)

__device__ __forceinline__ float wave32_sum(float v) {
    // wave32: 5 butterfly steps, no LDS, no barriers
#pragma unroll
    for (int off = 16; off > 0; off >>= 1)
        v += __shfl_xor(v, off, 32);
    return v;
}

__global__ __launch_bounds__(BLOCK) void gauss_loss_partial(
    const float* __restrict__ outputs,
    const int*   __restrict__ targets,
    float*       __restrict__ partials,
    int nrows)
{
    // Window weights: constant per offset; compiler folds to scalar v_s_exp_f32.
    const float inv_norm = 0.09973557010035818f; // 1/(sqrt(2*pi)*4)
    float w[WIN];
#pragma unroll
    for (int j = 0; j < WIN; ++j) {
        float d = (float)(j - BAND) * 0.25f;     // (k - y)/SIG, SIG=4
        w[j] = __expf(-0.5f * d * d) * inv_norm;
    }

    float acc = 0.0f;
    const int stride = gridDim.x * blockDim.x;
    for (int row = blockIdx.x * blockDim.x + threadIdx.x; row < nrows; row += stride) {
        const int y = targets[row];
        const float* __restrict__ rowp = outputs + (size_t)row * C_NUM;

        float Z = 0.0f;
        float s = 0.0f;
#pragma unroll
        for (int j = 0; j < WIN; ++j) {
            const int k = y + (j - BAND);
            const bool valid = (k >= 0) & (k < C_NUM);
            const int kc = min(max(k, 0), C_NUM - 1);
            // Non-temporal: outputs is 400MB streamed once; don't pollute L2.
            const float v  = __builtin_nontemporal_load(rowp + kc);
            const float vm = valid ? v : 0.0f;
            Z += vm;
            s += valid ? (w[j] * __logf(v)) : 0.0f;
        }
        // Z in ~(4e-4, 7): hardware reciprocal (1 ulp) beats the IEEE div chain.
        acc += s * __builtin_amdgcn_rcpf(Z);
    }

    // Intra-wave butterfly, then 8 wave partials through LDS.
    acc = wave32_sum(acc);

    __shared__ float red[WAVES_PER_BLOCK];
    const int lane = threadIdx.x & 31;
    const int wid  = threadIdx.x >> 5;
    if (lane == 0) red[wid] = acc;
    __syncthreads();
    if (threadIdx.x == 0) {
        float t = red[0];
#pragma unroll
        for (int i = 1; i < WAVES_PER_BLOCK; ++i) t += red[i];
        partials[blockIdx.x] = t;
    }
}

__global__ __launch_bounds__(BLOCK) void gauss_loss_final(
    const float* __restrict__ partials,
    int nparts,
    float* __restrict__ out,
    float neg_inv_b)
{
    float acc = 0.0f;
    for (int i = threadIdx.x; i < nparts; i += BLOCK)
        acc += partials[i];

    acc = wave32_sum(acc);

    __shared__ float red[WAVES_PER_BLOCK];
    const int lane = threadIdx.x & 31;
    const int wid  = threadIdx.x >> 5;
    if (lane == 0) red[wid] = acc;
    __syncthreads();
    if (threadIdx.x == 0) {
        float t = red[0];
#pragma unroll
        for (int i = 1; i < WAVES_PER_BLOCK; ++i) t += red[i];
        out[0] = t * neg_inv_b;
    }
}

extern "C" void kernel_launch(void* const* d_in, const int* in_sizes, int n_in,
                              void* d_out, int out_size, void* d_ws, size_t ws_size,
                              hipStream_t stream) {
    const float* outputs = (const float*)d_in[0];   // [B, 100] float32
    const int*   targets = (const int*)d_in[1];     // [B] int32
    float* out      = (float*)d_out;                // scalar loss
    float* partials = (float*)d_ws;                 // GRID floats of scratch

    const int nrows = in_sizes[1];                  // B

    // Every block writes its partial unconditionally -> no ws init needed,
    // no state carried across graph replays; fixed reduction order ->
    // bit-deterministic output.
    gauss_loss_partial<<<GRID, BLOCK, 0, stream>>>(outputs, targets, partials, nrows);
    gauss_loss_final<<<1, BLOCK, 0, stream>>>(partials, GRID, out,
                                              -1.0f / (float)nrows);
}